// IM2HT_60988535603686
// MI455X (gfx1250) — compile-verified
//
#include <hip/hip_runtime.h>
#include <stdint.h>

// ---------------------------------------------------------------------------
// Hough-vote scatter-add for MI455X (gfx1250).
//   out[bc, ht_idx[v]] += in[bc, im_idx[v]] * weight[v]   for bc in [0,128)
// Memory/atomic-bound (0.38 GFLOP vs tens of MB of L2-resident traffic);
// WMMA is inapplicable (no dense matmul). CDNA5 paths: async global->LDS
// vote staging (ASYNCcnt) + s_wait_asynccnt; channel-innermost layouts give
// coalesced gathers and coalesced f32 atomics that stay inside the 192 MB L2.
// ---------------------------------------------------------------------------

#define BC 128
constexpr int CHUNK = 1024;   // votes staged in LDS per block-iteration (12 KB)

#if defined(__has_builtin)
#  if __has_builtin(__builtin_amdgcn_global_load_async_to_lds_b128) && \
      __has_builtin(__builtin_amdgcn_global_load_async_to_lds_b32)
#    define HAVE_ASYNC_LDS 1
#  endif
#  if __has_builtin(__builtin_amdgcn_s_wait_asynccnt)
#    define HAVE_WAIT_ASYNC 1
#  endif
#endif

// Exact pointee types the builtins expect (per hipcc diagnostic):
//   b128: int __attribute__((vector_size(16)))  ;  b32: int
typedef int v4i_t __attribute__((vector_size(16)));
typedef __attribute__((address_space(1))) v4i_t* g_v4i_ptr;
typedef __attribute__((address_space(3))) v4i_t* l_v4i_ptr;
typedef __attribute__((address_space(1))) int*   g_i32_ptr;
typedef __attribute__((address_space(3))) int*   l_i32_ptr;

// Generic->AS1: global generic address == AS1 address (identity).
// Generic->AS3: low 32 bits of a generic LDS address are the LDS byte offset
// (ISA 10.2 aperture rules).
__device__ __forceinline__ g_v4i_ptr glb16(const void* p) {
  return (g_v4i_ptr)(uintptr_t)p;
}
__device__ __forceinline__ l_v4i_ptr lds16(void* p) {
  return (l_v4i_ptr)(uint32_t)(uintptr_t)p;
}
__device__ __forceinline__ g_i32_ptr glb4(const void* p) {
  return (g_i32_ptr)(uintptr_t)p;
}
__device__ __forceinline__ l_i32_ptr lds4(void* p) {
  return (l_i32_ptr)(uint32_t)(uintptr_t)p;
}

__device__ __forceinline__ void wait_async_lds() {
#if defined(HAVE_ASYNC_LDS)
#  if defined(HAVE_WAIT_ASYNC)
  __builtin_amdgcn_s_wait_asynccnt(0);
#  else
  asm volatile("s_wait_asynccnt 0" ::: "memory");
#  endif
#endif
}

// ---------------------------------------------------------------------------
// TRANS=true : in = ws_in[pix][128], out = acc[ht][128]   (coalesced fast path)
// TRANS=false: in = in[bc][pix],     out = d_out[bc][ht]  (fallback)
// blockDim = 256 (8 wave32): two 128-thread channel groups each take a
// contiguous half of the staged chunk.
// ---------------------------------------------------------------------------
template <bool TRANS>
__global__ __launch_bounds__(256) void hough_vote_kernel(
    const float* __restrict__ in, const int* __restrict__ im_idx,
    const int* __restrict__ ht_idx, const float* __restrict__ weight,
    float* __restrict__ out, int n, int im_size, int ht_size) {
  __shared__ __align__(16) int   s_im[CHUNK];
  __shared__ __align__(16) int   s_ht[CHUNK];
  __shared__ __align__(16) float s_w[CHUNK];

  const int t    = threadIdx.x;
  const int bc   = t & (BC - 1);
  const int vsub = t >> 7;  // 0 or 1: which half of the chunk

  for (int base = blockIdx.x * CHUNK; base < n; base += gridDim.x * CHUNK) {
    const int cnt   = min(CHUNK, n - base);
    const int full4 = cnt >> 2;  // number of 16-byte groups

#if defined(HAVE_ASYNC_LDS)
    if (t < full4) {
      __builtin_amdgcn_global_load_async_to_lds_b128(
          glb16(im_idx + base + 4 * t), lds16(&s_im[4 * t]), 0, 0);
      __builtin_amdgcn_global_load_async_to_lds_b128(
          glb16(ht_idx + base + 4 * t), lds16(&s_ht[4 * t]), 0, 0);
      __builtin_amdgcn_global_load_async_to_lds_b128(
          glb16(weight + base + 4 * t), lds16(&s_w[4 * t]), 0, 0);
    }
    if (t == 0) {  // ragged tail (< 4 votes)
      for (int j = 4 * full4; j < cnt; ++j) {
        __builtin_amdgcn_global_load_async_to_lds_b32(
            glb4(im_idx + base + j), lds4(&s_im[j]), 0, 0);
        __builtin_amdgcn_global_load_async_to_lds_b32(
            glb4(ht_idx + base + j), lds4(&s_ht[j]), 0, 0);
        __builtin_amdgcn_global_load_async_to_lds_b32(
            glb4(weight + base + j), lds4(&s_w[j]), 0, 0);
      }
    }
    wait_async_lds();
#else
    if (t < full4) {
      reinterpret_cast<int4*>(s_im)[t] =
          reinterpret_cast<const int4*>(im_idx + base)[t];
      reinterpret_cast<int4*>(s_ht)[t] =
          reinterpret_cast<const int4*>(ht_idx + base)[t];
      reinterpret_cast<float4*>(s_w)[t] =
          reinterpret_cast<const float4*>(weight + base)[t];
    }
    if (t == 0) {
      for (int j = 4 * full4; j < cnt; ++j) {
        s_im[j] = im_idx[base + j];
        s_ht[j] = ht_idx[base + j];
        s_w[j]  = weight[base + j];
      }
    }
#endif
    __syncthreads();

    const float* gin  = TRANS ? in  : in  + (size_t)bc * im_size;
    float*       gout = TRANS ? out : out + (size_t)bc * ht_size;

    const int half = (cnt + 1) >> 1;
    const int vbeg = vsub ? half : 0;
    const int vend = vsub ? cnt  : half;
#pragma unroll 4
    for (int v = vbeg; v < vend; ++v) {
      const int   im = s_im[v];          // LDS broadcast reads
      const int   ht = s_ht[v];
      const float wt = s_w[v];
      const float val = (TRANS ? gin[im * BC + bc] : gin[im]) * wt;
      atomicAdd(TRANS ? (gout + ht * BC + bc) : (gout + ht), val);
    }
    __syncthreads();  // protect LDS before next chunk is staged
  }
}

// Force emission here so the vote kernels lead the device assembly.
template __global__ void hough_vote_kernel<true>(
    const float* __restrict__, const int* __restrict__, const int* __restrict__,
    const float* __restrict__, float* __restrict__, int, int, int);
template __global__ void hough_vote_kernel<false>(
    const float* __restrict__, const int* __restrict__, const int* __restrict__,
    const float* __restrict__, float* __restrict__, int, int, int);

__global__ void hough_zero_kernel(float* __restrict__ p, long long n) {
  long long i = (long long)blockIdx.x * blockDim.x + threadIdx.x;
  const long long stride = (long long)gridDim.x * blockDim.x;
  for (; i < n; i += stride) p[i] = 0.0f;
}

// ws_in[pix][128] = in[bc][pix] ; coalesced reads along pix
__global__ void hough_tin_kernel(const float* __restrict__ in,
                                 float* __restrict__ ws, int im_size) {
  const int p  = blockIdx.x * blockDim.x + threadIdx.x;
  const int bc = blockIdx.y;
  if (p < im_size) ws[(size_t)p * BC + bc] = in[(size_t)bc * im_size + p];
}

// d_out[bc][ht] = acc[ht][128] ; coalesced writes along ht
__global__ void hough_tout_kernel(const float* __restrict__ acc,
                                  float* __restrict__ out, int ht_size) {
  const int ht = blockIdx.x * blockDim.x + threadIdx.x;
  const int bc = blockIdx.y;
  if (ht < ht_size) out[(size_t)bc * ht_size + ht] = acc[(size_t)ht * BC + bc];
}

extern "C" void kernel_launch(void* const* d_in, const int* in_sizes, int n_in,
                              void* d_out, int out_size, void* d_ws,
                              size_t ws_size, hipStream_t stream) {
  const float* in     = (const float*)d_in[0];
  const int*   im_idx = (const int*)d_in[1];
  const int*   ht_idx = (const int*)d_in[2];
  const float* weight = (const float*)d_in[3];
  float*       out    = (float*)d_out;

  const int n       = in_sizes[1];           // number of votes
  const int im_size = in_sizes[0] / BC;      // 16384
  const int ht_size = out_size / BC;         // 33120

  const size_t acc_bytes = (size_t)ht_size * BC * sizeof(float);
  const size_t in_bytes  = (size_t)im_size * BC * sizeof(float);
  const int    chunks    = (n + CHUNK - 1) / CHUNK;

  if (ws_size >= acc_bytes + in_bytes + 256) {
    float* acc   = (float*)d_ws;
    float* ws_in = (float*)((char*)d_ws + ((acc_bytes + 255) & ~(size_t)255));

    hough_zero_kernel<<<1024, 256, 0, stream>>>(acc, (long long)ht_size * BC);
    dim3 gin((im_size + 255) / 256, BC);
    hough_tin_kernel<<<gin, 256, 0, stream>>>(in, ws_in, im_size);
    hough_vote_kernel<true><<<chunks, 256, 0, stream>>>(
        ws_in, im_idx, ht_idx, weight, acc, n, im_size, ht_size);
    dim3 gout((ht_size + 255) / 256, BC);
    hough_tout_kernel<<<gout, 256, 0, stream>>>(acc, out, ht_size);
  } else {
    hough_zero_kernel<<<1024, 256, 0, stream>>>(out, (long long)out_size);
    hough_vote_kernel<false><<<chunks, 256, 0, stream>>>(
        in, im_idx, ht_idx, weight, out, n, im_size, ht_size);
  }
}